// CNNLayer_65867618452132
// MI455X (gfx1250) — compile-verified
//
#include <hip/hip_runtime.h>
#include <math.h>

// Problem constants (match reference)
#define BATCH 1024
#define H_    200
#define W_    32
#define COUT_ 16
#define TOPK_ 50

typedef float v2f __attribute__((ext_vector_type(2)));
typedef float v8f __attribute__((ext_vector_type(8)));

// LDS layout (bank-conflict padded; 64 banks x 4B on CDNA5)
#define XS_STRIDE      17            // 16 w-cols padded to 17
#define KEY_H_STRIDE   17            // 16 w-cols padded to 17
#define KEY_CO_STRIDE  3402          // 200*17 + 2 -> 8*3402 % 64 == 16 (half-wave store groups disjoint banks)
#define STG_T_STRIDE   17            // rank stride in stage tile
#define STG_CO_STRIDE  (TOPK_ * STG_T_STRIDE + 1)   // 851

__device__ __forceinline__ unsigned float_to_key(float v) {
  // order-preserving bijection f32 -> u32 (ascending), invertible
  unsigned u = __float_as_uint(v);
  return u ^ (unsigned)(((int)u >> 31) | 0x80000000);
}
__device__ __forceinline__ float key_to_float(unsigned k) {
  unsigned bits = (k & 0x80000000u) ? (k ^ 0x80000000u) : ~k;
  return __uint_as_float(bits);
}

__device__ __forceinline__ float fast_tanh(float v) {
#if __has_builtin(__builtin_amdgcn_tanhf)
  return __builtin_amdgcn_tanhf(v);          // gfx1250 V_TANH_F32 (TRANS32)
#elif __has_builtin(__builtin_amdgcn_tanh_f32)
  return __builtin_amdgcn_tanh_f32(v);
#else
  return tanhf(v);                           // OCML fallback
#endif
}

// Lane-xor exchange on the VALU (no LDS round trip):
//   MASK 1/2/4 -> DPP8 within 8-lane groups
//   MASK 8     -> v_permlane16_b32 (selector i^8 inside each 16-lane row)
//   MASK 16    -> v_permlanex16_b32 (identity selector, opposite row)
// Each path is feature-guarded with a __shfl_xor (LDS) fallback.
template <int MASK>
__device__ __forceinline__ unsigned lane_xor_t(unsigned v) {
  if constexpr (MASK < 8) {
#if __has_builtin(__builtin_amdgcn_mov_dpp8)
    constexpr unsigned sel = (unsigned)((0 ^ MASK)        | ((1 ^ MASK) << 3)  |
                                        ((2 ^ MASK) << 6) | ((3 ^ MASK) << 9)  |
                                        ((4 ^ MASK) << 12)| ((5 ^ MASK) << 15) |
                                        ((6 ^ MASK) << 18)| ((7 ^ MASK) << 21));
    return __builtin_amdgcn_mov_dpp8(v, sel);
#else
    return (unsigned)__shfl_xor((int)v, MASK, 32);
#endif
  } else if constexpr (MASK == 8) {
#if __has_builtin(__builtin_amdgcn_permlane16)
    return __builtin_amdgcn_permlane16(v, v, 0xFEDCBA98u, 0x76543210u, true, false);
#else
    return (unsigned)__shfl_xor((int)v, MASK, 32);
#endif
  } else {
#if __has_builtin(__builtin_amdgcn_permlanex16)
    return __builtin_amdgcn_permlanex16(v, v, 0x76543210u, 0xFEDCBA98u, true, false);
#else
    return (unsigned)__shfl_xor((int)v, MASK, 32);
#endif
  }
}
__device__ __forceinline__ unsigned lane_xor_rt(unsigned v, int lmask) {
  switch (lmask) {                      // lmask is an unrolled constant -> folds
    case 1:  return lane_xor_t<1>(v);
    case 2:  return lane_xor_t<2>(v);
    case 4:  return lane_xor_t<4>(v);
    case 8:  return lane_xor_t<8>(v);
    default: return lane_xor_t<16>(v);
  }
}

// Wave-wide bitonic sort of NSEQ independent 256-slot sequences, descending by
// slot index i = lane*8 + r  (lane-major: each lane owns 8 consecutive slots).
// Strides 1,2,4 are in-register pair-CEs; strides >=8 are lane-xor exchanges.
template <int NSEQ>
__device__ __forceinline__ void bitonic_sort_desc(unsigned key[NSEQ][8], int lane) {
#pragma unroll
  for (int k2 = 2; k2 <= 256; k2 <<= 1) {
#pragma unroll
    for (int j = 128; j >= 1; j >>= 1) {
      if (j >= k2) continue;
      if (j < 8) {
        // in-register compare-exchange: pair (r, r|j), slots i=lane*8+r
        const int rm = j;
#pragma unroll
        for (int r = 0; r < 8; ++r) {
          if ((r & rm) != 0) continue;
          const int r2 = r | rm;
          const int i  = lane * 8 + r;
          const bool desc = ((i & k2) == 0);
#pragma unroll
          for (int p = 0; p < NSEQ; ++p) {
            const unsigned a = key[p][r], c = key[p][r2];
            const unsigned mx = a > c ? a : c;
            const unsigned mn = a > c ? c : a;
            key[p][r]  = desc ? mx : mn;
            key[p][r2] = desc ? mn : mx;
          }
        }
      } else {
        // cross-lane compare-exchange: partner lane = lane ^ (j/8)
        const int lmask = j >> 3;
#pragma unroll
        for (int r = 0; r < 8; ++r) {
          const int i = lane * 8 + r;
          const bool desc     = ((i & k2) == 0);
          const bool upper    = (lane & lmask) != 0;     // == (i & j) != 0
          const bool take_max = (desc != upper);
#pragma unroll
          for (int p = 0; p < NSEQ; ++p) {
            const unsigned other = lane_xor_rt(key[p][r], lmask);
            const unsigned mine  = key[p][r];
            const unsigned mx = mine > other ? mine : other;
            const unsigned mn = mine > other ? other : mine;
            key[p][r] = take_max ? mx : mn;
          }
        }
      }
    }
  }
}

__global__ __launch_bounds__(512) void cnn_conv_tanh_topk(
    const float* __restrict__ x,    // [B,1,200,32]
    const float* __restrict__ cw,   // [16,1,3,1]
    const float* __restrict__ cb,   // [16]
    float* __restrict__ out)        // [B,16,50,32]
{
  __shared__ float    xs[202 * XS_STRIDE];            // 13.7 KB: x tile + zero halo rows 0 and 201
  __shared__ unsigned keys[COUT_ * KEY_CO_STRIDE];    // 217.7 KB: tanh(conv) as monotone keys
  __shared__ float    stage[COUT_ * STG_CO_STRIDE];   // 54.5 KB: [co][rank][wl] result staging

  const int tid  = (int)threadIdx.x;
  const int wv   = tid >> 5;        // wave id 0..15
  const int lane = tid & 31;
  const int b    = (int)blockIdx.x >> 1;
  const int w0   = ((int)blockIdx.x & 1) * 16;

  // ---------------- Phase 0: stage x[b][:, w0:w0+16] into LDS (+ zero halo rows) ----
  for (int i = tid; i < 202 * 16; i += 512) {
    const int hr = i >> 4;          // padded row 0..201; h = hr-1
    const int wl = i & 15;
    const int h  = hr - 1;
    float v = 0.0f;
    if (h >= 0 && h < H_) v = x[(b * H_ + h) * W_ + w0 + wl];
    xs[hr * XS_STRIDE + wl] = v;
  }

  // A fragment: 16x4 f32 weights (K=3 padded with 0), branch-free per ISA layout:
  // lanes 0-15: M=lane, v0=K0, v1=K1 ; lanes 16-31: M=lane-16, v0=K2, v1=K3.
  const int m  = lane & 15;
  const int hi = lane >> 4;         // 0 for lanes 0-15, 1 for lanes 16-31
  v2f afrag;
  afrag.x = cw[m * 3 + 2 * hi];                 // K0 (lo) / K2 (hi)
  afrag.y = hi ? 0.0f : cw[m * 3 + 1];          // K1 (lo) / K3 pad (hi)

  // C fragment: bias broadcast. D/C layout: vgpr r -> row (r + 8*hi), lane%16 -> col.
  v8f cfrag;
#pragma unroll
  for (int r = 0; r < 8; ++r) cfrag[r] = cb[r + 8 * hi];

  __syncthreads();

  // ---------------- Phase 1: conv+bias via V_WMMA_F32_16X16X4_F32, tanh, key store --
  // Wave wv covers h = wv, wv+16, ... (uniform per wave -> EXEC all ones at WMMA).
  for (int h = wv; h < H_; h += 16) {
    // B fragment 4x16: rows K0..K3 = x[h-1], x[h], x[h+1], 0 ; cols = 16 w.
    // Branch-free: hi lanes take rows (h+2, 0); LDS row 0 is the all-zero halo.
    const int rowA = h + 2 * hi;                // x[h-1] (lo) / x[h+1] (hi)
    const int rowB = hi ? 0 : (h + 1);          // x[h]   (lo) / zeros  (hi)
    v2f bfrag;
    bfrag.x = xs[rowA * XS_STRIDE + m];
    bfrag.y = xs[rowB * XS_STRIDE + m];

    v8f d = __builtin_amdgcn_wmma_f32_16x16x4_f32(
        false, afrag, false, bfrag, (short)0, cfrag, false, false);
#pragma unroll
    for (int r = 0; r < 8; ++r) {
      const float v = fast_tanh(d[r]);
      const int  co = r + 8 * hi;
      keys[co * KEY_CO_STRIDE + h * KEY_H_STRIDE + m] = float_to_key(v);
    }
  }

  __syncthreads();

  // ---------------- Phase 2: per-(co,w) wave-wide bitonic top-50 (descending) -------
  // Wave wv owns co = wv; processes 2 w-columns per pass (ILP).
  // Lane-major slots: lane owns h = lane*8 .. lane*8+7 (slots beyond 199 -> key 0).
  const int co = wv;
  for (int wl = 0; wl < 16; wl += 2) {
    unsigned key[2][8];
#pragma unroll
    for (int r = 0; r < 8; ++r) {
      const int h = lane * 8 + r;
      const int base = co * KEY_CO_STRIDE + h * KEY_H_STRIDE;
      key[0][r] = (h < H_) ? keys[base + wl]     : 0u;
      key[1][r] = (h < H_) ? keys[base + wl + 1] : 0u;
    }

    bitonic_sort_desc<2>(key, lane);

    // rank t lives at lane = t/8, reg = t%8 ; lanes 0..6 hold ranks 0..55
#pragma unroll
    for (int r = 0; r < 8; ++r) {
      const int t = lane * 8 + r;
      if (t < TOPK_) {
        stage[co * STG_CO_STRIDE + t * STG_T_STRIDE + wl]     = key_to_float(key[0][r]);
        stage[co * STG_CO_STRIDE + t * STG_T_STRIDE + wl + 1] = key_to_float(key[1][r]);
      }
    }
  }

  __syncthreads();

  // ---------------- Phase 3: coalesced float4 store of the staged [16,50,16] tile ---
  for (int i4 = tid * 4; i4 < COUT_ * TOPK_ * 16; i4 += 512 * 4) {
    const int oco = i4 / (TOPK_ * 16);
    const int rem = i4 % (TOPK_ * 16);
    const int t   = rem >> 4;
    const int wl  = rem & 15;       // multiple of 4
    const int sbase = oco * STG_CO_STRIDE + t * STG_T_STRIDE + wl;
    float4 v;
    v.x = stage[sbase + 0];
    v.y = stage[sbase + 1];
    v.z = stage[sbase + 2];
    v.w = stage[sbase + 3];
    *(float4*)&out[((long)(b * COUT_ + oco) * TOPK_ + t) * W_ + w0 + wl] = v;
  }
}

extern "C" void kernel_launch(void* const* d_in, const int* in_sizes, int n_in,
                              void* d_out, int out_size, void* d_ws, size_t ws_size,
                              hipStream_t stream) {
  const float* x  = (const float*)d_in[0];   // [1024,1,200,32] f32
  const float* cw = (const float*)d_in[1];   // [16,1,3,1]      f32
  const float* cb = (const float*)d_in[2];   // [16]            f32
  float* out = (float*)d_out;                // [1024,16,50,32] f32

  (void)in_sizes; (void)n_in; (void)out_size; (void)d_ws; (void)ws_size;

  dim3 grid(BATCH * 2);   // (batch, w-half)
  dim3 block(512);        // 16 wave32 waves
  cnn_conv_tanh_topk<<<grid, block, 0, stream>>>(x, cw, cb, out);
}